// Batch_Dynamic_Atten_Autoencoder_28166395527746
// MI455X (gfx1250) — compile-verified
//
#include <hip/hip_runtime.h>
#include <hip/hip_bf16.h>

// ---------------------------------------------------------------------------
// Problem constants (from reference)
// ---------------------------------------------------------------------------
#define NN      10000
#define IN_F    512
#define OUT_F   64
#define NE      160000
#define MT      625          // NN / 16  row tiles
#define GY      79           // ceil(MT / 8) — 8 waves (16-row tiles) per block
#define KC      16           // K chunk staged in LDS (divides 64, 512, 10000)

typedef __attribute__((ext_vector_type(2))) float v2f;
typedef __attribute__((ext_vector_type(8))) float v8f;

// ---------------------------------------------------------------------------
// Helpers
// ---------------------------------------------------------------------------
__device__ __forceinline__ float warp_sum(float v) {
  #pragma unroll
  for (int off = 16; off > 0; off >>= 1) v += __shfl_down(v, off, 32);
  return v;
}
__device__ __forceinline__ float sigmoidf(float x) { return 1.f / (1.f + expf(-x)); }

// order-preserving float<->uint encoding for atomicMax on floats
__device__ __forceinline__ unsigned f2ord(float f) {
  unsigned u = __float_as_uint(f);
  return (u & 0x80000000u) ? ~u : (u | 0x80000000u);
}
__device__ __forceinline__ float ord2f(unsigned e) {
  unsigned u = (e & 0x80000000u) ? (e & 0x7fffffffu) : ~e;
  return __uint_as_float(u);
}
#define NEG_INF_ENC 0x007FFFFFu   // f2ord(-inf)

// ---------------------------------------------------------------------------
// Fill kernels (no hipMemsetAsync: keep graph-capture simple)
// ---------------------------------------------------------------------------
__global__ void k_fill_f32(float* p, float v, int n) {
  int i = blockIdx.x * 256 + threadIdx.x;
  if (i < n) p[i] = v;
}
__global__ void k_fill_u32(unsigned* p, unsigned v, int n) {
  int i = blockIdx.x * 256 + threadIdx.x;
  if (i < n) p[i] = v;
}

// ---------------------------------------------------------------------------
// WMMA fp32 GEMM: C[M,N] = A[M,K] @ B[K,N], row-major, M=16*Mtiles, K%KC==0.
// blockDim = 256 (8 waves). Each wave owns one 16-row tile; all 8 waves share
// a 16*NT-column B panel staged in LDS (double-buffered, KC rows per chunk).
// B pairs stored interleaved [k/2][n][k&1] so a fragment is one ds_load_b64.
// grid = (N/(16*NT), ceil(Mtiles/8)).
// ---------------------------------------------------------------------------
template <int NT>
__global__ void k_gemm_wmma(const float* __restrict__ A, int lda,
                            const float* __restrict__ B, int ldb,
                            float* __restrict__ C, int ldc,
                            int Mtiles, int K) {
  constexpr int WN    = 16 * NT;        // panel width
  constexpr int CHUNK = KC * WN;        // floats per staged chunk
  constexpr int PER_T = (CHUNK + 255) / 256;
  __shared__ float sb[2][CHUNK];

  const int tid  = threadIdx.x;
  const int wave = tid >> 5;
  const int lane = tid & 31;
  int tileM = blockIdx.y * 8 + wave;
  const bool active = (tileM < Mtiles);
  if (!active) tileM = Mtiles - 1;      // keep loads in-bounds, stay in barriers
  const int l    = lane & 15;
  const int half = lane >> 4;
  const int colBase = blockIdx.x * WN;

  const float* Arow = A + (size_t)(tileM * 16 + l) * lda + 2 * half;
  const float* Bg   = B + colBase;

  float rg[PER_T];
  auto load_chunk = [&](int k0) {
    #pragma unroll
    for (int i = 0; i < PER_T; ++i) {
      int idx = tid + i * 256;
      if (idx < CHUNK) {
        int row = idx / WN, col = idx - row * WN;
        rg[i] = Bg[(size_t)(k0 + row) * ldb + col];
      }
    }
  };
  auto store_chunk = [&](float* dstb) {
    #pragma unroll
    for (int i = 0; i < PER_T; ++i) {
      int idx = tid + i * 256;
      if (idx < CHUNK) {
        int row = idx / WN, col = idx - row * WN;
        dstb[((row >> 1) * WN + col) * 2 + (row & 1)] = rg[i];
      }
    }
  };

  const v8f zero = {0.f, 0.f, 0.f, 0.f, 0.f, 0.f, 0.f, 0.f};
  v8f acc[NT];
  #pragma unroll
  for (int t = 0; t < NT; ++t) acc[t] = zero;

  load_chunk(0);
  store_chunk(sb[0]);
  __syncthreads();

  const int nchunk = K / KC;
  int buf = 0;
  for (int c = 0; c < nchunk; ++c) {
    if (c + 1 < nchunk) load_chunk((c + 1) * KC);   // issue next-chunk globals
    const float* sbuf = sb[buf];
    const float* Ac   = Arow + (size_t)c * KC;
    #pragma unroll
    for (int kk = 0; kk < KC; kk += 4) {
      v2f a = *reinterpret_cast<const v2f*>(Ac + kk);          // A[m][k..k+1]
      const int prow = (kk + 2 * half) >> 1;                   // pair row
      const float* sp = sbuf + (prow * WN + l) * 2;
      #pragma unroll
      for (int t = 0; t < NT; ++t) {
        v2f b = *reinterpret_cast<const v2f*>(sp + t * 32);    // B[k][n],B[k+1][n]
        acc[t] = __builtin_amdgcn_wmma_f32_16x16x4_f32(
            false, a, false, b, (short)0, acc[t], false, false);
      }
    }
    if (c + 1 < nchunk) store_chunk(sb[buf ^ 1]);   // fill other buffer
    __syncthreads();
    buf ^= 1;
  }

  if (active) {
    float* Crow = C + (size_t)(tileM * 16 + half * 8) * ldc + colBase + l;
    #pragma unroll
    for (int t = 0; t < NT; ++t)
      #pragma unroll
      for (int r = 0; r < 8; ++r)
        Crow[(size_t)r * ldc + t * 16] = acc[t][r];
  }
}

// ---------------------------------------------------------------------------
// Edge attention (GATv2, segment softmax), 3 passes. One wave per edge for the
// wide passes, one thread per edge for the exp/denom pass.
// ---------------------------------------------------------------------------
__global__ void k_edge_scores(const float* __restrict__ xl, const float* __restrict__ xr,
                              const float* __restrict__ att,
                              const int* __restrict__ src, const int* __restrict__ dst,
                              int ne, int W, float* __restrict__ e_out,
                              unsigned* __restrict__ m_enc) {
  int edge = blockIdx.x * 8 + (threadIdx.x >> 5);
  int lane = threadIdx.x & 31;
  if (edge >= ne) return;
  int s = src[edge], d = dst[edge];
  const float* pl = xl + (size_t)s * W;
  const float* pr = xr + (size_t)d * W;
  float acc = 0.f;
  for (int f = lane; f < W; f += 32) {
    float v = pl[f] + pr[f];
    v = (v > 0.f) ? v : 0.2f * v;        // leaky_relu, slope 0.2
    acc += v * att[f];
  }
  acc = warp_sum(acc);
  if (lane == 0) {
    e_out[edge] = acc;
    atomicMax(m_enc + d, f2ord(acc));
  }
}

__global__ void k_edge_softmax(const float* __restrict__ e, const int* __restrict__ dst,
                               int ne, const unsigned* __restrict__ m_enc,
                               float* __restrict__ ex, float* __restrict__ denom) {
  int k = blockIdx.x * 256 + threadIdx.x;
  if (k >= ne) return;
  int d = dst[k];
  float v = expf(e[k] - ord2f(m_enc[d]));
  ex[k] = v;
  atomicAdd(denom + d, v);
}

__global__ void k_edge_aggregate(const float* __restrict__ xl,
                                 const int* __restrict__ src, const int* __restrict__ dst,
                                 const float* __restrict__ ex, const float* __restrict__ denom,
                                 int ne, int W, float* __restrict__ c, int ldc, int col_off) {
  int edge = blockIdx.x * 8 + (threadIdx.x >> 5);
  int lane = threadIdx.x & 31;
  if (edge >= ne) return;
  int s = src[edge], d = dst[edge];
  float alpha = ex[edge] / (denom[d] + 1e-16f);
  const float* pl = xl + (size_t)s * W;
  float* pc = c + (size_t)d * ldc + col_off;
  for (int f = lane; f < W; f += 32) atomicAdd(pc + f, pl[f] * alpha);
}

// ---------------------------------------------------------------------------
// Epilogue helpers
// ---------------------------------------------------------------------------
// z_pack[N,128] -> z (raw, to output), emb = relu(z), emb_a = relu(z_a)
__global__ void k_extract(const float* __restrict__ zp, float* __restrict__ z_out,
                          float* __restrict__ emb, float* __restrict__ emb_a, int n64) {
  int i = blockIdx.x * 256 + threadIdx.x;
  if (i >= n64) return;
  int row = i >> 6, col = i & 63;
  float a = zp[(size_t)row * 128 + col];
  float b = zp[(size_t)row * 128 + 64 + col];
  z_out[i] = a;
  emb[i]   = fmaxf(a, 0.f);
  emb_a[i] = fmaxf(b, 0.f);
}

// B_r[N,144]: cols 0-63 = emb, 64-127 = emb_a, col 128 = 1 (row-sum trick), rest 0
__global__ void k_build_Br(const float* __restrict__ emb, const float* __restrict__ emb_a,
                           float* __restrict__ Br, int total) {
  int i = blockIdx.x * 256 + threadIdx.x;
  if (i >= total) return;
  int row = i / 144, col = i - row * 144;
  float v;
  if (col < 64)       v = emb[(size_t)row * 64 + col];
  else if (col < 128) v = emb_a[(size_t)row * 64 + col - 64];
  else                v = (col == 128) ? 1.f : 0.f;
  Br[i] = v;
}

// R[N,144] = graph_neigh @ B_r  ->  g = sigmoid(normalize(vsum/row_sum))
__global__ void k_readout(const float* __restrict__ R, float* __restrict__ g,
                          float* __restrict__ g_a, int n) {
  int node = blockIdx.x * 8 + (threadIdx.x >> 5);
  int lane = threadIdx.x & 31;
  if (node >= n) return;
  const float* r = R + (size_t)node * 144;
  float inv_rs = 1.f / r[128];
  float a0 = r[lane] * inv_rs,      a1 = r[lane + 32] * inv_rs;
  float b0 = r[64 + lane] * inv_rs, b1 = r[96 + lane] * inv_rs;
  float na = warp_sum(a0 * a0 + a1 * a1);
  float nb = warp_sum(b0 * b0 + b1 * b1);
  na = __shfl(na, 0, 32);
  nb = __shfl(nb, 0, 32);
  float ia = 1.f / fmaxf(sqrtf(na), 1e-12f);
  float ib = 1.f / fmaxf(sqrtf(nb), 1e-12f);
  g[(size_t)node * 64 + lane]        = sigmoidf(a0 * ia);
  g[(size_t)node * 64 + lane + 32]   = sigmoidf(a1 * ia);
  g_a[(size_t)node * 64 + lane]      = sigmoidf(b0 * ib);
  g_a[(size_t)node * 64 + lane + 32] = sigmoidf(b1 * ib);
}

// u1 = emb@w_d, u2 = emb_a@w_d already computed; 4 bilinear dots per node
__global__ void k_scores(const float* __restrict__ u1, const float* __restrict__ u2,
                         const float* __restrict__ g, const float* __restrict__ g_a,
                         const float* __restrict__ b_d,
                         float* __restrict__ ret, float* __restrict__ ret_a, int n) {
  int node = blockIdx.x * 8 + (threadIdx.x >> 5);
  int lane = threadIdx.x & 31;
  if (node >= n) return;
  const float* p1 = u1 + (size_t)node * 64;
  const float* p2 = u2 + (size_t)node * 64;
  const float* pg = g + (size_t)node * 64;
  const float* pa = g_a + (size_t)node * 64;
  float s1 = 0, s2 = 0, s1a = 0, s2a = 0;
  #pragma unroll
  for (int f = lane; f < 64; f += 32) {
    float ue = p1[f], ua = p2[f], gv = pg[f], gav = pa[f];
    s1 += ue * gv; s2 += ua * gv; s1a += ua * gav; s2a += ue * gav;
  }
  s1 = warp_sum(s1); s2 = warp_sum(s2); s1a = warp_sum(s1a); s2a = warp_sum(s2a);
  if (lane == 0) {
    float b = b_d[0];
    ret[(size_t)node * 2 + 0]   = sigmoidf(s1 + b);
    ret[(size_t)node * 2 + 1]   = sigmoidf(s2 + b);
    ret_a[(size_t)node * 2 + 0] = sigmoidf(s1a + b);
    ret_a[(size_t)node * 2 + 1] = sigmoidf(s2a + b);
  }
}

// ---------------------------------------------------------------------------
// Host orchestration
// ---------------------------------------------------------------------------
extern "C" void kernel_launch(void* const* d_in, const int* in_sizes, int n_in,
                              void* d_out, int out_size, void* d_ws, size_t ws_size,
                              hipStream_t stream) {
  const float* feat    = (const float*)d_in[0];
  const float* feat_a  = (const float*)d_in[1];
  const float* gneigh  = (const float*)d_in[2];
  const float* adj     = (const float*)d_in[3];
  const int*   eidx    = (const int*)d_in[4];
  const float* w_l1    = (const float*)d_in[5];
  const float* w_r1    = (const float*)d_in[6];
  const float* att1    = (const float*)d_in[7];
  const float* w_l2    = (const float*)d_in[8];
  const float* w_r2    = (const float*)d_in[9];
  const float* att2    = (const float*)d_in[10];
  const float* w_d     = (const float*)d_in[11];
  const float* b_d     = (const float*)d_in[12];
  const int* src = eidx;
  const int* dst = eidx + NE;

  float* out_z   = (float*)d_out;                       // [N,64]
  float* out_h   = out_z + (size_t)NN * OUT_F;          // [N,512]
  float* out_ret = out_h + (size_t)NN * IN_F;           // [N,2]
  float* out_reta= out_ret + (size_t)NN * 2;            // [N,2]

  // workspace layout (floats)
  float* ws = (float*)d_ws;
  float*    x_l1   = ws;                         // N*64
  float*    x_r1   = x_l1 + (size_t)NN * 64;
  float*    x_l1a  = x_r1 + (size_t)NN * 64;
  float*    x_r1a  = x_l1a + (size_t)NN * 64;
  float*    e_buf  = x_r1a + (size_t)NN * 64;    // E
  float*    ex_buf = e_buf + NE;                 // E
  unsigned* m_enc  = (unsigned*)(ex_buf + NE);   // N
  float*    denom  = (float*)(m_enc + NN);       // N
  float*    cc     = denom + NN;                 // N*128  (conv1 out: feat|feat_a)
  float*    z_pack = cc + (size_t)NN * 128;      // N*128  (z|z_a)
  float*    x_l2   = z_pack + (size_t)NN * 128;  // N*512
  float*    x_r2   = x_l2 + (size_t)NN * 512;    // N*512
  float*    c2     = x_r2 + (size_t)NN * 512;    // N*512  (conv2 out)
  float*    emb    = c2 + (size_t)NN * 512;      // N*64
  float*    emb_a  = emb + (size_t)NN * 64;      // N*64
  float*    Br     = emb_a + (size_t)NN * 64;    // N*144
  float*    Rr     = Br + (size_t)NN * 144;      // N*144
  float*    gbuf   = Rr + (size_t)NN * 144;      // N*64
  float*    gabuf  = gbuf + (size_t)NN * 64;     // N*64
  float*    u1     = gabuf + (size_t)NN * 64;    // N*64
  float*    u2     = u1 + (size_t)NN * 64;       // N*64

  dim3 blk(256);
  const int EB = (NE + 7) / 8;            // wave-per-edge blocks
  const int ET = (NE + 255) / 256;        // thread-per-edge blocks
  #define FILLG(n) dim3(((n) + 255) / 256)

  // --- Stage A: layer-1 projections (M=10000, K=512, N=64) -----------------
  k_gemm_wmma<4><<<dim3(1, GY), blk, 0, stream>>>(feat,   IN_F, w_l1, OUT_F, x_l1,  OUT_F, MT, IN_F);
  k_gemm_wmma<4><<<dim3(1, GY), blk, 0, stream>>>(feat,   IN_F, w_r1, OUT_F, x_r1,  OUT_F, MT, IN_F);
  k_gemm_wmma<4><<<dim3(1, GY), blk, 0, stream>>>(feat_a, IN_F, w_l1, OUT_F, x_l1a, OUT_F, MT, IN_F);
  k_gemm_wmma<4><<<dim3(1, GY), blk, 0, stream>>>(feat_a, IN_F, w_r1, OUT_F, x_r1a, OUT_F, MT, IN_F);

  // --- Stage B: layer-1 edge softmax + aggregation into cc[N,128] ----------
  k_fill_f32<<<FILLG(NN * 128), blk, 0, stream>>>(cc, 0.f, NN * 128);
  // feat branch -> cc cols 0..63
  k_fill_u32<<<FILLG(NN), blk, 0, stream>>>(m_enc, NEG_INF_ENC, NN);
  k_fill_f32<<<FILLG(NN), blk, 0, stream>>>(denom, 0.f, NN);
  k_edge_scores   <<<dim3(EB), blk, 0, stream>>>(x_l1, x_r1, att1, src, dst, NE, OUT_F, e_buf, m_enc);
  k_edge_softmax  <<<dim3(ET), blk, 0, stream>>>(e_buf, dst, NE, m_enc, ex_buf, denom);
  k_edge_aggregate<<<dim3(EB), blk, 0, stream>>>(x_l1, src, dst, ex_buf, denom, NE, OUT_F, cc, 128, 0);
  // feat_a branch -> cc cols 64..127
  k_fill_u32<<<FILLG(NN), blk, 0, stream>>>(m_enc, NEG_INF_ENC, NN);
  k_fill_f32<<<FILLG(NN), blk, 0, stream>>>(denom, 0.f, NN);
  k_edge_scores   <<<dim3(EB), blk, 0, stream>>>(x_l1a, x_r1a, att1, src, dst, NE, OUT_F, e_buf, m_enc);
  k_edge_softmax  <<<dim3(ET), blk, 0, stream>>>(e_buf, dst, NE, m_enc, ex_buf, denom);
  k_edge_aggregate<<<dim3(EB), blk, 0, stream>>>(x_l1a, src, dst, ex_buf, denom, NE, OUT_F, cc, 128, 64);

  // --- Stage C: z_pack = adj @ cc  (M=10000, K=10000, N=128) ---------------
  k_gemm_wmma<4><<<dim3(2, GY), blk, 0, stream>>>(adj, NN, cc, 128, z_pack, 128, MT, NN);

  // --- Stage D: split z (output), emb = relu(z), emb_a = relu(z_a) ---------
  k_extract<<<FILLG(NN * 64), blk, 0, stream>>>(z_pack, out_z, emb, emb_a, NN * 64);

  // --- Stage E: layer-2 projections (A = z in z_pack, lda=128, K=64, N=512)
  k_gemm_wmma<4><<<dim3(8, GY), blk, 0, stream>>>(z_pack, 128, w_l2, IN_F, x_l2, IN_F, MT, OUT_F);
  k_gemm_wmma<4><<<dim3(8, GY), blk, 0, stream>>>(z_pack, 128, w_r2, IN_F, x_r2, IN_F, MT, OUT_F);

  // --- Stage F: layer-2 edge softmax + aggregation into c2[N,512] ----------
  k_fill_f32<<<FILLG(NN * 512), blk, 0, stream>>>(c2, 0.f, NN * 512);
  k_fill_u32<<<FILLG(NN), blk, 0, stream>>>(m_enc, NEG_INF_ENC, NN);
  k_fill_f32<<<FILLG(NN), blk, 0, stream>>>(denom, 0.f, NN);
  k_edge_scores   <<<dim3(EB), blk, 0, stream>>>(x_l2, x_r2, att2, src, dst, NE, IN_F, e_buf, m_enc);
  k_edge_softmax  <<<dim3(ET), blk, 0, stream>>>(e_buf, dst, NE, m_enc, ex_buf, denom);
  k_edge_aggregate<<<dim3(EB), blk, 0, stream>>>(x_l2, src, dst, ex_buf, denom, NE, IN_F, c2, 512, 0);

  // --- Stage G: h = adj @ c2  (M=10000, K=10000, N=512) --------------------
  k_gemm_wmma<4><<<dim3(8, GY), blk, 0, stream>>>(adj, NN, c2, 512, out_h, 512, MT, NN);

  // --- Stage H: readout — R = graph_neigh @ [emb|emb_a|1|0...] -------------
  k_build_Br<<<FILLG(NN * 144), blk, 0, stream>>>(emb, emb_a, Br, NN * 144);
  k_gemm_wmma<3><<<dim3(3, GY), blk, 0, stream>>>(gneigh, NN, Br, 144, Rr, 144, MT, NN);
  k_readout<<<dim3((NN + 7) / 8), blk, 0, stream>>>(Rr, gbuf, gabuf, NN);

  // --- Stage I: bilinear scores -------------------------------------------
  k_gemm_wmma<4><<<dim3(1, GY), blk, 0, stream>>>(emb,   OUT_F, w_d, OUT_F, u1, OUT_F, MT, OUT_F);
  k_gemm_wmma<4><<<dim3(1, GY), blk, 0, stream>>>(emb_a, OUT_F, w_d, OUT_F, u2, OUT_F, MT, OUT_F);
  k_scores<<<dim3((NN + 7) / 8), blk, 0, stream>>>(u1, u2, gbuf, gabuf, b_d, out_ret, out_reta, NN);

  (void)in_sizes; (void)n_in; (void)out_size; (void)ws_size;
}